// SCAB_Encoder_Batch_16449724744101
// MI455X (gfx1250) — compile-verified
//
#include <hip/hip_runtime.h>
#include <cstddef>
#include <cstdint>

typedef float v2f __attribute__((ext_vector_type(2)));
typedef float v4f __attribute__((ext_vector_type(4)));
typedef float v8f __attribute__((ext_vector_type(8)));
typedef unsigned int v4u __attribute__((ext_vector_type(4)));
typedef int v4i __attribute__((ext_vector_type(4)));
typedef int v8i __attribute__((ext_vector_type(8)));

// D = A(16x4,f32) * B(4x16,f32) + C(16x16,f32)  -- full-precision CDNA5 matrix op
#define WMMA4(A_, B_, C_) \
  __builtin_amdgcn_wmma_f32_16x16x4_f32(false, (A_), false, (B_), (short)0, (C_), false, false)

// ---------------------------------------------------------------------------
// Tensor Data Mover: 2D tile (dword elements) global -> LDS.
// Descriptor packing per CDNA5 ISA:
//   g0: [1:0]=count=1, [63:32]=lds_addr, [120:64]=global_addr, [127:126]=type(2)
//   g1: [17:16]=data_size(2=4B), [20]=pad_en, [24:22]=pad_interval, [31:25]=pad_amount,
//       [79:48]=tensor_dim0, [111:80]=tensor_dim1, [127:112]=tile_dim0,
//       [143:128]=tile_dim1, [207:160]=tensor_dim0_stride
//   groups 2/3 zero (2D tile).
// pad_interval code: 0=2,1=4,2=8,3=16,4=32,... dwords; pad_amount code: n -> n+1 dwords
// ---------------------------------------------------------------------------
__device__ __forceinline__ void tdm_load_2d(unsigned lds_addr, const void* gaddr,
                                            unsigned tile_d0, unsigned tile_d1,
                                            unsigned stride_dw, unsigned pad_en,
                                            unsigned pad_interval, unsigned pad_amount)
{
  const unsigned long long ga = (unsigned long long)(size_t)gaddr;
  v4u g0;
  g0.x = 1u;                                                  // count=1, user mode
  g0.y = lds_addr;                                            // LDS byte address
  g0.z = (unsigned)(ga & 0xffffffffu);                        // global_addr[31:0]
  g0.w = (unsigned)((ga >> 32) & 0x01ffffffu) | (2u << 30);   // global_addr[56:32], type=2
  v8i g1;
  g1[0] = (int)((2u << 16) | (pad_en << 20) | (pad_interval << 22) | (pad_amount << 25));
  g1[1] = (int)((tile_d0 & 0xffffu) << 16);                   // tensor_dim0 = tile_d0 (lo16)
  g1[2] = (int)(((tile_d0 >> 16) & 0xffffu) | ((tile_d1 & 0xffffu) << 16)); // dim0 hi, dim1 lo
  g1[3] = (int)(((tile_d1 >> 16) & 0xffffu) | ((tile_d0 & 0xffffu) << 16)); // dim1 hi, tile_dim0
  g1[4] = (int)(tile_d1 & 0xffffu);                           // tile_dim1; tile_dim2=0
  g1[5] = (int)stride_dw;                                     // tensor_dim0_stride[31:0]
  g1[6] = 0;                                                  // stride hi bits / dim1_stride lo
  g1[7] = 0;
  const v4i z4 = {0, 0, 0, 0};
#if defined(__clang_major__) && __clang_major__ >= 23
  const v8i z8 = {0, 0, 0, 0, 0, 0, 0, 0};
  __builtin_amdgcn_tensor_load_to_lds(g0, g1, z4, z4, z8, 0);
#else
  __builtin_amdgcn_tensor_load_to_lds(g0, g1, z4, z4, 0);
#endif
}

__device__ __forceinline__ unsigned lds_off(const void* p)
{
  return (unsigned)(size_t)p;   // LDS aperture: flat addr[31:0] = LDS byte offset
}

// ---------------------------------------------------------------------------
// GEMM: C[b] = A[b](MxK) @ W(KxN) + bias  [+ReLU] [+Res[b]]
// block = 256 thr = 8 waves (2x4), wave tile 32x32 (2x2 wmma tiles), K step 16
// TDM double-buffered LDS staging (A padded to 20 floats/row via TDM pad).
// ---------------------------------------------------------------------------
__global__ __launch_bounds__(256) void gemm_kernel(
    const float* __restrict__ A, const float* __restrict__ Wt,
    const float* __restrict__ bias, const float* __restrict__ Res,
    float* __restrict__ C, int M, int N, int K, int relu)
{
  const int b    = blockIdx.z;
  const int n0   = blockIdx.x * 128;
  const int m0   = blockIdx.y * 64;
  const int tid  = threadIdx.x;
  const int wave = tid >> 5, lane = tid & 31;
  const int wm = wave >> 2, wn = wave & 3;   // 2 x 4 wave grid
  const int lo = lane & 15, hi = lane >> 4;

  __shared__ float sA[2][64 * 20];           // 64 x 16, TDM-padded to stride 20
  __shared__ float sB[2][16 * 128];

  const float* Ab = A + (size_t)b * M * K + (size_t)m0 * K;
  v8f acc00 = {}, acc01 = {}, acc10 = {}, acc11 = {};

  if (wave == 0) {
    tdm_load_2d(lds_off(&sA[0][0]), Ab,            16, 64, (unsigned)K, 1, 3, 3);
    tdm_load_2d(lds_off(&sB[0][0]), Wt + n0,      128, 16, (unsigned)N, 0, 0, 0);
  }

  int cur = 0;
  for (int k0 = 0; k0 < K; k0 += 16) {
    const bool more = (k0 + 16) < K;
    if (wave == 0) {
      if (more) {
        tdm_load_2d(lds_off(&sA[cur ^ 1][0]), Ab + (k0 + 16), 16, 64, (unsigned)K, 1, 3, 3);
        tdm_load_2d(lds_off(&sB[cur ^ 1][0]), Wt + (size_t)(k0 + 16) * N + n0,
                    128, 16, (unsigned)N, 0, 0, 0);
        __builtin_amdgcn_s_wait_tensorcnt(2);   // current stage's 2 TDMs complete
      } else {
        __builtin_amdgcn_s_wait_tensorcnt(0);
      }
    }
    __syncthreads();
    const float* pA = &sA[cur][0];
    const float* pB = &sB[cur][0];
#pragma unroll
    for (int kk = 0; kk < 4; ++kk) {
      const int kc = kk * 4 + 2 * hi;           // A layout: lane = m + 16*(k>>1), reg = k&1
      v2f a0 = *(const v2f*)&pA[(wm * 32 + lo) * 20 + kc];
      v2f a1 = *(const v2f*)&pA[(wm * 32 + 16 + lo) * 20 + kc];
      const int nb = wn * 32 + lo;              // B layout: lane = n + 16*(k>>1), reg = k&1
      v2f b0 = { pB[kc * 128 + nb],      pB[(kc + 1) * 128 + nb] };
      v2f b1 = { pB[kc * 128 + nb + 16], pB[(kc + 1) * 128 + nb + 16] };
      acc00 = WMMA4(a0, b0, acc00);
      acc01 = WMMA4(a0, b1, acc01);
      acc10 = WMMA4(a1, b0, acc10);
      acc11 = WMMA4(a1, b1, acc11);
    }
    __syncthreads();                            // all reads done before buffer reuse
    cur ^= 1;
  }

  const float* Rb = Res ? Res + (size_t)b * M * N : nullptr;
  float* Cb = C + (size_t)b * M * N;
  v8f accs[2][2] = { { acc00, acc01 }, { acc10, acc11 } };
#pragma unroll
  for (int mt = 0; mt < 2; ++mt) {
#pragma unroll
    for (int nt = 0; nt < 2; ++nt) {
      const int col = n0 + wn * 32 + nt * 16 + lo;
      const float bv = bias[col];
#pragma unroll
      for (int r = 0; r < 8; ++r) {             // C layout: vgpr r -> rows r (lanes 0-15), r+8 (16-31)
        const int row = m0 + wm * 32 + mt * 16 + r + 8 * hi;
        float v = accs[mt][nt][r] + bv;
        if (relu) v = fmaxf(v, 0.f);
        if (Rb) v += Rb[(size_t)row * N + col];
        Cb[(size_t)row * N + col] = v;
      }
    }
  }
}

// ---------------------------------------------------------------------------
// Flash attention: per (b, h, 128 q-rows). D=256, H=8, dh=32.
// scores scaled by 1/sqrt(D)=1/16 (reference semantics), key-masked softmax.
// K/V tiles (16x32, TDM-padded to 34) double-buffered via TDM.
// ---------------------------------------------------------------------------
__global__ __launch_bounds__(256) void attn_kernel(
    const float* __restrict__ Qp, const float* __restrict__ Kp,
    const float* __restrict__ Vp, const int* __restrict__ lenK,
    float* __restrict__ O, int Lq, int Lk)
{
  const int b = blockIdx.z, h = blockIdx.y;
  const int tid = threadIdx.x, wave = tid >> 5, lane = tid & 31;
  const int lo = lane & 15, hi = lane >> 4;
  const int q0 = blockIdx.x * 128 + wave * 16;

  __shared__ float sK[2][16 * 34];    // 16 keys x 32 dh, TDM-padded to stride 34
  __shared__ float sV[2][16 * 34];
  __shared__ float sP[8][16 * 18];    // per-wave P transpose scratch (16x16, pad 18)

  const float* Qb = Qp + ((size_t)b * Lq) * 256 + h * 32;
  const float* Kb = Kp + ((size_t)b * Lk) * 256 + h * 32;
  const float* Vb = Vp + ((size_t)b * Lk) * 256 + h * 32;

  v2f qf[8];                          // 16 q-rows x dh=32 A-frags, pre-scaled
#pragma unroll
  for (int j = 0; j < 8; ++j) {
    const int c = j * 4 + 2 * hi;
    v2f t = *(const v2f*)(Qb + (size_t)(q0 + lo) * 256 + c);
    qf[j] = t * 0.0625f;
  }

  v8f accO0 = {}, accO1 = {};
  float mrow[8], lrow[8];
#pragma unroll
  for (int r = 0; r < 8; ++r) { mrow[r] = -3.0e38f; lrow[r] = 0.f; }
  const int nlen = lenK[b];

  if (wave == 0) {                    // pad: 2 dwords after every 32 -> row stride 34
    tdm_load_2d(lds_off(&sK[0][0]), Kb, 32, 16, 256u, 1, 4, 1);
    tdm_load_2d(lds_off(&sV[0][0]), Vb, 32, 16, 256u, 1, 4, 1);
  }

  int cur = 0;
  for (int kt = 0; kt < Lk; kt += 16) {
    const bool more = (kt + 16) < Lk;
    if (wave == 0) {
      if (more) {
        tdm_load_2d(lds_off(&sK[cur ^ 1][0]), Kb + (size_t)(kt + 16) * 256, 32, 16, 256u, 1, 4, 1);
        tdm_load_2d(lds_off(&sV[cur ^ 1][0]), Vb + (size_t)(kt + 16) * 256, 32, 16, 256u, 1, 4, 1);
        __builtin_amdgcn_s_wait_tensorcnt(2);
      } else {
        __builtin_amdgcn_s_wait_tensorcnt(0);
      }
    }
    __syncthreads();

    // S(16q x 16k) = Q * K^T  (8 WMMAs over dh=32)
    const float* pK = &sK[cur][0];
    v8f s = {};
#pragma unroll
    for (int j = 0; j < 8; ++j) {
      const int c = j * 4 + 2 * hi;
      v2f kb = *(const v2f*)&pK[lo * 34 + c];
      s = WMMA4(qf[j], kb, s);
    }

    const bool valid = (kt + lo) < nlen;   // this lane's key column
#pragma unroll
    for (int r = 0; r < 8; ++r) {
      float sv = valid ? s[r] : -1.0e30f;
      float mx = sv;                        // row max across the 16-lane half
      mx = fmaxf(mx, __shfl_xor(mx, 1, 32));
      mx = fmaxf(mx, __shfl_xor(mx, 2, 32));
      mx = fmaxf(mx, __shfl_xor(mx, 4, 32));
      mx = fmaxf(mx, __shfl_xor(mx, 8, 32));
      const float mnew = fmaxf(mrow[r], mx);
      const float scl  = __expf(mrow[r] - mnew);
      const float pv   = __expf(sv - mnew);
      float ps = pv;
      ps += __shfl_xor(ps, 1, 32);
      ps += __shfl_xor(ps, 2, 32);
      ps += __shfl_xor(ps, 4, 32);
      ps += __shfl_xor(ps, 8, 32);
      lrow[r] = lrow[r] * scl + ps;
      mrow[r] = mnew;
      accO0[r] *= scl;
      accO1[r] *= scl;
      sP[wave][(r + 8 * hi) * 18 + lo] = pv;   // C-layout -> row-major scratch
    }
    __syncthreads();

    // O(16q x 32dh) += P(16x16) @ V(16x32)  (4 k-steps x 2 n-tiles)
    const float* pV = &sV[cur][0];
#pragma unroll
    for (int jj = 0; jj < 4; ++jj) {
      const int kc = jj * 4 + 2 * hi;
      v2f pa = *(const v2f*)&sP[wave][lo * 18 + kc];
      v2f vb0 = { pV[kc * 34 + lo],      pV[(kc + 1) * 34 + lo] };
      v2f vb1 = { pV[kc * 34 + 16 + lo], pV[(kc + 1) * 34 + 16 + lo] };
      accO0 = WMMA4(pa, vb0, accO0);
      accO1 = WMMA4(pa, vb1, accO1);
    }
    __syncthreads();                        // V reads done before buffer reuse
    cur ^= 1;
  }

  float* Ob = O + ((size_t)b * Lq) * 256 + h * 32;
#pragma unroll
  for (int r = 0; r < 8; ++r) {
    const int row = q0 + r + 8 * hi;
    const float inv = 1.f / lrow[r];
    Ob[(size_t)row * 256 + lo]      = accO0[r] * inv;
    Ob[(size_t)row * 256 + 16 + lo] = accO1[r] * inv;
  }
}

// ---------------------------------------------------------------------------
// LayerNorm over D=256: out = LN(in [+res]) * g + be ; one wave per row
// ---------------------------------------------------------------------------
__global__ __launch_bounds__(256) void ln_kernel(
    const float* __restrict__ in, const float* __restrict__ res,
    const float* __restrict__ g, const float* __restrict__ be,
    float* __restrict__ out)
{
  const int wave = threadIdx.x >> 5, lane = threadIdx.x & 31;
  const int row = blockIdx.x * 8 + wave;
  const float* p = in + (size_t)row * 256 + lane * 8;
  v4f a = *(const v4f*)p;
  v4f c = *(const v4f*)(p + 4);
  if (res) {
    const float* q = res + (size_t)row * 256 + lane * 8;
    a += *(const v4f*)q;
    c += *(const v4f*)(q + 4);
  }
  float s = a.x + a.y + a.z + a.w + c.x + c.y + c.z + c.w;
  for (int m = 1; m <= 16; m <<= 1) s += __shfl_xor(s, m, 32);
  const float mean = s * (1.f / 256.f);
  v4f da = a - mean, dc = c - mean;
  float vs = da.x * da.x + da.y * da.y + da.z * da.z + da.w * da.w
           + dc.x * dc.x + dc.y * dc.y + dc.z * dc.z + dc.w * dc.w;
  for (int m = 1; m <= 16; m <<= 1) vs += __shfl_xor(vs, m, 32);
  const float inv = rsqrtf(vs * (1.f / 256.f) + 1e-5f);
  const int cb = lane * 8;
  float* o = out + (size_t)row * 256 + cb;
#pragma unroll
  for (int i = 0; i < 4; ++i) o[i]     = da[i] * inv * g[cb + i]     + be[cb + i];
#pragma unroll
  for (int i = 0; i < 4; ++i) o[4 + i] = dc[i] * inv * g[cb + 4 + i] + be[cb + 4 + i];
}

// out[idx] = X[idx] * (l < len[b])
__global__ __launch_bounds__(256) void catmask_kernel(
    const float* __restrict__ X, const int* __restrict__ len, float* __restrict__ out)
{
  const size_t i = (size_t)blockIdx.x * 256 + threadIdx.x;
  const int b = (int)(i / (512 * 256));
  const int l = (int)((i / 256) % 512);
  out[i] = (l < len[b]) ? X[i] : 0.f;
}

// out[d] += (1/B) * (1/len_b) * sum_{l<len_b} XA[b,l,d]
__global__ __launch_bounds__(256) void aggr_kernel(
    const float* __restrict__ XA, const int* __restrict__ len,
    float* __restrict__ out, float invB)
{
  const int b = blockIdx.x, d = threadIdx.x;
  const int n = len[b];
  float acc = 0.f;
  for (int l = 0; l < n; ++l) {
    if (l + 8 < n) __builtin_prefetch(&XA[((size_t)b * 512 + l + 8) * 256 + d], 0, 1);
    acc += XA[((size_t)b * 512 + l) * 256 + d];
  }
  atomicAdd(&out[d], acc * invB / (float)n);
}

__global__ __launch_bounds__(256) void zero2_kernel(float* a, float* b)
{
  a[threadIdx.x] = 0.f;
  b[threadIdx.x] = 0.f;
}

// ---------------------------------------------------------------------------
// Host orchestration
// ---------------------------------------------------------------------------
struct MabP {
  const float *qw, *qb, *kw, *kb, *vw, *vb, *g0, *be0;
  const float *o1w, *o1b, *o2w, *o2b, *g1, *be1;
  bool ff;
};

extern "C" void kernel_launch(void* const* d_in, const int* in_sizes, int n_in,
                              void* d_out, int out_size, void* d_ws, size_t ws_size,
                              hipStream_t stream)
{
  (void)in_sizes; (void)n_in; (void)out_size; (void)ws_size;
  constexpr int B = 32, L = 512;
  const size_t SZ = (size_t)B * L * 256;   // 4,194,304 floats per activation tensor

  // ---- unpack inputs (setup_inputs() dict insertion order, recursive; _lin = w,b)
  int idx = 0;
  auto nf = [&]() { return (const float*)d_in[idx++]; };
  const float* x_in = nf();
  const float* y_in = nf();
  auto readMab = [&](bool ff) {
    MabP m{};
    m.qw = nf(); m.qb = nf(); m.kw = nf(); m.kb = nf(); m.vw = nf(); m.vb = nf();
    m.g0 = nf(); m.be0 = nf();
    if (ff) { m.o1w = nf(); m.o1b = nf(); m.o2w = nf(); m.o2b = nf(); m.g1 = nf(); m.be1 = nf(); }
    m.ff = ff;
    return m;
  };
  MabP selfp[2];
  selfp[0] = readMab(true);
  selfp[1] = readMab(true);
  MabP xy0[2], xy1[2], yx0[2], yx1[2];
  for (int i = 0; i < 2; ++i) { xy0[i] = readMab(false); xy1[i] = readMab(true); }
  for (int i = 0; i < 2; ++i) { yx0[i] = readMab(false); yx1[i] = readMab(true); }
  const float* axl1w = nf(); const float* axl1b = nf();
  const float* axl2w = nf(); const float* axl2b = nf();
  const float* ayl1w = nf(); const float* ayl1b = nf();
  const float* ayl2w = nf(); const float* ayl2b = nf();
  const int* len_x = (const int*)d_in[idx++];
  const int* len_y = (const int*)d_in[idx++];

  // ---- workspace layout (11 x 16MB)
  float* W  = (float*)d_ws;
  float* X  = W + 0 * SZ;
  float* Y  = W + 1 * SZ;
  float* XN = W + 2 * SZ;
  float* YN = W + 3 * SZ;
  float* X2 = W + 4 * SZ;
  float* Y2 = W + 5 * SZ;
  float* Tb = W + 6 * SZ;
  float* Qb = W + 7 * SZ;
  float* Kb = W + 8 * SZ;
  float* Vb = W + 9 * SZ;
  float* Ob = W + 10 * SZ;
  float* Hb = Qb;   // [B,512,1024] aliases dead Q/K/V/O during the FF stage

  hipMemcpyAsync(X, x_in, SZ * sizeof(float), hipMemcpyDeviceToDevice, stream);
  hipMemcpyAsync(Y, y_in, SZ * sizeof(float), hipMemcpyDeviceToDevice, stream);

  auto gemm = [&](const float* A, const float* Wt, const float* bias, const float* Res,
                  float* C, int M, int N, int K, int relu) {
    dim3 g(N / 128, M / 64, B);
    gemm_kernel<<<g, 256, 0, stream>>>(A, Wt, bias, Res, C, M, N, K, relu);
  };
  const int rows = B * L;
  auto mab = [&](const float* Qin, const float* Kin, const int* lenK, const MabP& p,
                 float* outBuf) {
    gemm(Qin, p.qw, p.qb, nullptr, Qb, L, 256, 256, 0);
    gemm(Kin, p.kw, p.kb, nullptr, Kb, L, 256, 256, 0);
    gemm(Kin, p.vw, p.vb, nullptr, Vb, L, 256, 256, 0);
    dim3 ga(L / 128, 8, B);
    attn_kernel<<<ga, 256, 0, stream>>>(Qb, Kb, Vb, lenK, Ob, L, L);
    if (p.ff) {
      ln_kernel<<<rows / 8, 256, 0, stream>>>(Ob, Qin, p.g0, p.be0, Tb);
      gemm(Tb, p.o1w, p.o1b, nullptr, Hb, L, 1024, 256, 1);
      gemm(Hb, p.o2w, p.o2b, Tb, outBuf, L, 256, 1024, 0);
      ln_kernel<<<rows / 8, 256, 0, stream>>>(outBuf, nullptr, p.g1, p.be1, outBuf);
    } else {
      ln_kernel<<<rows / 8, 256, 0, stream>>>(Ob, Qin, p.g0, p.be0, outBuf);
    }
  };

  // ---- self-attention stack
  for (int i = 0; i < 2; ++i) {
    mab(X, X, len_x, selfp[i], XN); { float* t = X; X = XN; XN = t; }
    mab(Y, Y, len_y, selfp[i], YN); { float* t = Y; Y = YN; YN = t; }
  }
  // ---- cross-attention stack (both updates read pre-update x,y)
  for (int i = 0; i < 2; ++i) {
    mab(X,  X, len_x, xy0[i], XN);   // tx
    mab(XN, Y, len_y, xy1[i], X2);   // x2 = mab1(tx, y)
    mab(Y,  Y, len_y, yx0[i], YN);   // ty
    mab(YN, X, len_x, yx1[i], Y2);   // y2 = mab1(ty, x)
    { float* t = X; X = X2; X2 = t; }
    { float* t = Y; Y = Y2; Y2 = t; }
  }

  // ---- outputs: [y_aggr(256), x_cat(SZ), y_cat(SZ), x_aggr(256)]
  float* out    = (float*)d_out;
  float* y_aggr = out;
  float* x_cat  = out + 256;
  float* y_cat  = out + 256 + SZ;
  float* x_aggr = out + 256 + 2 * SZ;
  zero2_kernel<<<1, 256, 0, stream>>>(y_aggr, x_aggr);

  gemm(X, axl1w, axl1b, nullptr, Qb, L, 256, 256, 1);
  gemm(Qb, axl2w, axl2b, X, Kb, L, 256, 256, 0);
  aggr_kernel<<<B, 256, 0, stream>>>(Kb, len_x, x_aggr, 1.f / B);

  gemm(Y, ayl1w, ayl1b, nullptr, Qb, L, 256, 256, 1);
  gemm(Qb, ayl2w, ayl2b, Y, Kb, L, 256, 256, 0);
  aggr_kernel<<<B, 256, 0, stream>>>(Kb, len_y, y_aggr, 1.f / B);

  catmask_kernel<<<(unsigned)(SZ / 256), 256, 0, stream>>>(X, len_x, x_cat);
  catmask_kernel<<<(unsigned)(SZ / 256), 256, 0, stream>>>(Y, len_y, y_cat);
}